// Network_82394652607110
// MI455X (gfx1250) — compile-verified
//
#include <hip/hip_runtime.h>
#include <hip/hip_bf16.h>

typedef __attribute__((ext_vector_type(2))) float v2f;
typedef __attribute__((ext_vector_type(4))) float v4f;
typedef __attribute__((ext_vector_type(8))) float v8f;

#define U_DIM 5
#define T_LEN 1024
#define B_LEN 512
#define D_DIM 64
#define PX_PITCH 16   // padded row of px in floats (cols 0..4 = p1, 5..9 = p2)

// ---------------------------------------------------------------------------
// Phase 1: px[m, 0..9] = tx[m, :] @ kernel + bias   (m = b*T + t, M = 524288)
// One wave computes one 16(M) x 16(N) tile via 16x V_WMMA_F32_16X16X4_F32.
// Memory bound: tx read non-temporally (keep px hot in L2 for the scan).
// ---------------------------------------------------------------------------
__global__ __launch_bounds__(256) void px_gemm_wmma(
    const float* __restrict__ tx,     // (M, 64)
    const float* __restrict__ kern,   // (64, 10)
    const float* __restrict__ bias,   // (10)
    float* __restrict__ px,           // (M, 16) padded
    int M)
{
  const int lane = threadIdx.x & 31;
  const int wave = blockIdx.x * (blockDim.x >> 5) + (threadIdx.x >> 5);
  const int tile_m = wave * 16;
  if (tile_m >= M) return;                    // wave-uniform

  const int n = lane & 15;                    // N index (col)
  const int u = lane >> 4;                    // 0: lanes 0-15, 1: lanes 16-31

  // Preload all B fragments (K = 64 -> 16 chunks of 4). kernel is tiny / L2 hot.
  v2f bfrag[16];
#pragma unroll
  for (int kk = 0; kk < 16; ++kk) {
    const int k0 = kk * 4 + 2 * u;            // fragment k = k0 + v
    float b0 = 0.f, b1 = 0.f;
    if (n < 10) {
      b0 = kern[k0 * 10 + n];
      b1 = kern[(k0 + 1) * 10 + n];
    }
    bfrag[kk] = (v2f){b0, b1};
  }

  const float bn = (n < 10) ? bias[n] : 0.f;
  v8f c;
#pragma unroll
  for (int r = 0; r < 8; ++r) c[r] = bn;      // C initialized with bias

  // A fragment: row = tile_m + n (here n doubles as row-in-tile), k = 2u + v
  const float* arow = tx + (size_t)(tile_m + n) * D_DIM + 2 * u;
#pragma unroll
  for (int kk = 0; kk < 16; ++kk) {
    v2f a = __builtin_nontemporal_load(
        reinterpret_cast<const v2f*>(arow + kk * 4));
    // 8 args: (neg_a, A, neg_b, B, c_mod, C, reuse_a, reuse_b)
    c = __builtin_amdgcn_wmma_f32_16x16x4_f32(
        false, a, false, bfrag[kk], (short)0, c, false, false);
  }

  // D store: VGPR r -> row tile_m + r + 8u, col n. Two packed 64B rows/store.
  float* prow = px + (size_t)(tile_m + 8 * u) * PX_PITCH + n;
#pragma unroll
  for (int r = 0; r < 8; ++r) prow[r * PX_PITCH] = c[r];
}

// ---------------------------------------------------------------------------
// Phase 2: sequential gated scan over T, one thread per batch element, then
// fc + softmax tail. Weights register-resident; sigmoid/tanh via HW exp2/rcp;
// next-step px row preloaded + prefetched to hide L2 latency.
// ---------------------------------------------------------------------------
__global__ __launch_bounds__(256) void gated_scan(
    const float* __restrict__ rec,    // (5, 10): [:, :5]=Rf, [:, 5:]=Rh
    const float* __restrict__ fcw,    // (5, 4)
    const float* __restrict__ fcb,    // (4)
    const float* __restrict__ px,     // (B*T, 16) padded
    float* __restrict__ out)          // (B, 4)
{
  const int b = blockIdx.x * blockDim.x + threadIdx.x;
  if (b >= B_LEN) return;

  float Rf[U_DIM][U_DIM], Rh[U_DIM][U_DIM];
#pragma unroll
  for (int i = 0; i < U_DIM; ++i)
#pragma unroll
    for (int j = 0; j < U_DIM; ++j) {
      Rf[i][j] = rec[i * 10 + j];
      Rh[i][j] = rec[i * 10 + U_DIM + j];
    }

  float h[U_DIM] = {0.f, 0.f, 0.f, 0.f, 0.f};
  const float* row = px + (size_t)b * T_LEN * PX_PITCH;
  const float LOG2E = 1.4426950408889634f;

  // software pipeline: q* hold step t, n* hold step t+1
  v4f q0 = *reinterpret_cast<const v4f*>(row);
  v4f q1 = *reinterpret_cast<const v4f*>(row + 4);
  v2f q2 = *reinterpret_cast<const v2f*>(row + 8);

  for (int t = 0; t < T_LEN; ++t) {
    const int tp = (t + 8 < T_LEN) ? (t + 8) : (T_LEN - 1);
    __builtin_prefetch(row + (size_t)tp * PX_PITCH, 0, 3);

    v4f n0 = q0; v4f n1 = q1; v2f n2 = q2;
    if (t + 1 < T_LEN) {                      // uniform branch
      const float* nr = row + (size_t)(t + 1) * PX_PITCH;
      n0 = *reinterpret_cast<const v4f*>(nr);
      n1 = *reinterpret_cast<const v4f*>(nr + 4);
      n2 = *reinterpret_cast<const v2f*>(nr + 8);
    }

    const float p[10] = {q0[0], q0[1], q0[2], q0[3], q1[0],
                         q1[1], q1[2], q1[3], q2[0], q2[1]};

    float v1[U_DIM], g[U_DIM], v2[U_DIM];
#pragma unroll
    for (int j = 0; j < U_DIM; ++j) {         // v1 = sigmoid(p1 + h @ Rf)
      float s = p[j];
#pragma unroll
      for (int i = 0; i < U_DIM; ++i) s = __builtin_fmaf(h[i], Rf[i][j], s);
      const float e = __builtin_amdgcn_exp2f(-s * LOG2E);
      v1[j] = __builtin_amdgcn_rcpf(1.f + e);
    }
#pragma unroll
    for (int i = 0; i < U_DIM; ++i) g[i] = h[i] * v1[i];
#pragma unroll
    for (int j = 0; j < U_DIM; ++j) {         // v2 = tanh(p2 + (h*v1) @ Rh)
      float s = p[U_DIM + j];
#pragma unroll
      for (int i = 0; i < U_DIM; ++i) s = __builtin_fmaf(g[i], Rh[i][j], s);
      const float e = __builtin_amdgcn_exp2f(-2.f * s * LOG2E);
      const float sg = __builtin_amdgcn_rcpf(1.f + e);
      v2[j] = 2.f * sg - 1.f;                 // tanh(x) = 2*sigmoid(2x) - 1
    }
#pragma unroll
    for (int j = 0; j < U_DIM; ++j)           // h = (1-v1)*h + v1*v2
      h[j] = (1.f - v1[j]) * h[j] + v1[j] * v2[j];

    q0 = n0; q1 = n1; q2 = n2;
  }

  // fc + softmax tail
  float logits[4];
#pragma unroll
  for (int o = 0; o < 4; ++o) {
    float s = fcb[o];
#pragma unroll
    for (int i = 0; i < U_DIM; ++i) s = __builtin_fmaf(h[i], fcw[i * 4 + o], s);
    logits[o] = s;
  }
  float mx = logits[0];
#pragma unroll
  for (int o = 1; o < 4; ++o) mx = fmaxf(mx, logits[o]);
  float e[4], sum = 0.f;
#pragma unroll
  for (int o = 0; o < 4; ++o) {
    e[o] = __builtin_amdgcn_exp2f((logits[o] - mx) * LOG2E);
    sum += e[o];
  }
  const float inv = 1.f / sum;
#pragma unroll
  for (int o = 0; o < 4; ++o) out[b * 4 + o] = e[o] * inv;
}

extern "C" void kernel_launch(void* const* d_in, const int* in_sizes, int n_in,
                              void* d_out, int out_size, void* d_ws, size_t ws_size,
                              hipStream_t stream) {
  const float* tx   = (const float*)d_in[0];   // (512, 1024, 64)
  const float* kern = (const float*)d_in[1];   // (64, 10)
  const float* rec  = (const float*)d_in[2];   // (5, 10)
  const float* bias = (const float*)d_in[3];   // (10)
  const float* fcw  = (const float*)d_in[4];   // (5, 4)
  const float* fcb  = (const float*)d_in[5];   // (4)
  float* out = (float*)d_out;                  // (512, 4)
  float* px  = (float*)d_ws;                   // (B*T, 16) = 33.6 MB scratch

  const int M = B_LEN * T_LEN;                 // 524288 rows
  const int waves = M / 16;                    // one 16x16 tile per wave
  const int blocks = waves / 8;                // 256 threads = 8 waves/block

  px_gemm_wmma<<<blocks, 256, 0, stream>>>(tx, kern, bias, px, M);
  gated_scan<<<B_LEN / 256, 256, 0, stream>>>(rec, fcw, fcb, px, out);
}